// Top2Router_26611617366084
// MI455X (gfx1250) — compile-verified
//
#include <hip/hip_runtime.h>
#include <hip/hip_bf16.h>

typedef float v4f __attribute__((ext_vector_type(4)));

#define NUM_EXPERTS 64

// Per-token record, 8 ints (32 B, 16B-aligned):
//   [0]=idx1 [1]=idx2 [2]=rank1(-1 if dropped) [3]=rank2 [4]=w1 bits [5]=w2 bits [6,7]=pad
// ---------------------------------------------------------------------------
// Kernel 1: per-token softmax + top1/top2 selection. One wave32 per token.
// ---------------------------------------------------------------------------
__global__ void router_top2_kernel(const float* __restrict__ in,
                                   int* __restrict__ rec, int s) {
    int wave = (int)((blockIdx.x * blockDim.x + threadIdx.x) >> 5);
    int lane = threadIdx.x & 31;
    if (wave >= s) return;

    const float* row = in + (size_t)wave * NUM_EXPERTS;
    float x0 = row[lane];
    float x1 = row[lane + 32];

    // fp32 max-subtracted softmax (matches jax.nn.softmax)
    float m = fmaxf(x0, x1);
    for (int o = 16; o; o >>= 1) m = fmaxf(m, __shfl_xor(m, o, 32));
    float p0 = expf(x0 - m);
    float p1 = expf(x1 - m);
    float sum = p0 + p1;
    for (int o = 16; o; o >>= 1) sum += __shfl_xor(sum, o, 32);
    float inv = 1.0f / sum;
    p0 *= inv; p1 *= inv;

    // top1 argmax, first-index tie-break (jnp.argmax semantics)
    float v; int ix;
    if (p0 >= p1) { v = p0; ix = lane; } else { v = p1; ix = lane + 32; }
    for (int o = 16; o; o >>= 1) {
        float ov = __shfl_xor(v, o, 32);
        int   oi = __shfl_xor(ix, o, 32);
        if (ov > v || (ov == v && oi < ix)) { v = ov; ix = oi; }
    }
    int t1 = ix; float wt1 = v;

    // top2: exclude top1 (probs >= 0, so -1 acts as -inf)
    float q0 = (lane == t1)      ? -1.0f : p0;
    float q1 = (lane + 32 == t1) ? -1.0f : p1;
    if (q0 >= q1) { v = q0; ix = lane; } else { v = q1; ix = lane + 32; }
    for (int o = 16; o; o >>= 1) {
        float ov = __shfl_xor(v, o, 32);
        int   oi = __shfl_xor(ix, o, 32);
        if (ov > v || (ov == v && oi < ix)) { v = ov; ix = oi; }
    }

    if (lane == 0) {
        int* rp = rec + (size_t)wave * 8;
        int2 ii; ii.x = t1; ii.y = ix;
        *(int2*)rp = ii;                                   // idx1, idx2
        float2 ww; ww.x = wt1; ww.y = v;
        *(float2*)(rp + 4) = ww;                           // w1, w2
    }
}

// ---------------------------------------------------------------------------
// Kernel 2: per-expert exclusive rank (cumsum-1) over tokens, capacity mask.
// Single block, 256-token chunks. Double-buffered CDNA5 async global->LDS
// staging (ASYNCcnt pipeline: issue chunk c+1, s_wait_asynccnt 1 => chunk c
// resident). In-chunk ranks via 6 ballots (expert id is 6 bits) + per-wave
// LDS histograms. Pass 2 inherits pass-1 totals => rank2 offset by count1[e],
// exactly matching the reference.
// ---------------------------------------------------------------------------
__global__ void rank_scan_kernel(int* __restrict__ rec, int s, int cap) {
    __shared__ int sbuf[2][256];
    __shared__ int hist[8 * NUM_EXPERTS];
    __shared__ int base[NUM_EXPERTS];
    const int tid  = threadIdx.x;
    const int lane = tid & 31;
    const int wv   = tid >> 5;

    if (tid < NUM_EXPERTS) base[tid] = 0;
    __syncthreads();

    const int chunks = s >> 8;
    for (int pass = 0; pass < 2; ++pass) {
        // prologue: request chunk 0 (idx field stride is 8 ints per token)
        {
            unsigned int       lds = (unsigned int)(unsigned long long)&sbuf[0][tid];
            unsigned long long ga  = (unsigned long long)(const void*)(rec + (size_t)tid * 8 + pass);
            asm volatile("global_load_async_to_lds_b32 %0, %1, off"
                         :: "v"(lds), "v"(ga) : "memory");
        }
        for (int c = 0; c < chunks; ++c) {
            if (c + 1 < chunks) {
                int ntok = ((c + 1) << 8) + tid;
                unsigned int       lds = (unsigned int)(unsigned long long)&sbuf[(c + 1) & 1][tid];
                unsigned long long ga  = (unsigned long long)(const void*)(rec + (size_t)ntok * 8 + pass);
                asm volatile("global_load_async_to_lds_b32 %0, %1, off"
                             :: "v"(lds), "v"(ga) : "memory");
                asm volatile("s_wait_asynccnt 0x1" ::: "memory");  // chunk c done (in-order)
            } else {
                asm volatile("s_wait_asynccnt 0x0" ::: "memory");
            }
            __syncthreads();                       // chunk c resident for all waves

            int e = sbuf[c & 1][tid];

            // lanes in my wave with the same expert: AND of 6 ballot planes
            unsigned m = 0xffffffffu;
#pragma unroll
            for (int b = 0; b < 6; ++b) {
                unsigned bb = __builtin_amdgcn_ballot_w32((e >> b) & 1);
                m &= (((e >> b) & 1) ? bb : ~bb);
            }
            unsigned ltmask     = (1u << lane) - 1u;
            int      prevInWave = __builtin_popcount(m & ltmask);
            bool     leader     = (m & ltmask) == 0u;
            int      cntInWave  = __builtin_popcount(m);

            hist[tid] = 0; hist[tid + 256] = 0;
            __syncthreads();
            if (leader) hist[wv * NUM_EXPERTS + e] = cntInWave;
            __syncthreads();

            int prior = 0;
#pragma unroll
            for (int w = 0; w < 8; ++w)
                if (w < wv) prior += hist[w * NUM_EXPERTS + e];

            int rk  = base[e] + prior + prevInWave;
            int tok = (c << 8) + tid;
            rec[(size_t)tok * 8 + 2 + pass] = (rk < cap) ? rk : -1;
            __syncthreads();                       // everyone done reading base
            if (tid < NUM_EXPERTS) {
                int tot = 0;
#pragma unroll
                for (int w = 0; w < 8; ++w) tot += hist[w * NUM_EXPERTS + tid];
                base[tid] += tot;                  // carries into pass 2 => count1 offset
            }
            __syncthreads();
        }
    }
}

// ---------------------------------------------------------------------------
// Kernel 3: fused zero-fill + scatter — the roofline pass (1 GB of NT b128
// stores). One thread owns one float4 of cb_weight + matching float4 of
// sec_mask. Token index is wave-uniform (4096 threads per token), so the
// 32-byte record is fetched through a readfirstlane'd (SGPR) address ->
// scalar loads, no redundant per-lane VMEM.
// ---------------------------------------------------------------------------
__global__ void fill_scatter_kernel(const int* __restrict__ rec,
                                    float* __restrict__ out,
                                    unsigned int nvec,   // N/4
                                    long long N,         // s*e*cap
                                    int tptShift, int cgShift) {
    unsigned int i = blockIdx.x * 256u + threadIdx.x;
    if (i >= nvec) return;

    int      t   = (int)(i >> tptShift);
    unsigned rem = i & ((1u << tptShift) - 1u);
    int      e   = (int)(rem >> cgShift);
    int      cg  = (int)(rem & ((1u << cgShift) - 1u));
    int      c0  = cg << 2;

    int ts = __builtin_amdgcn_readfirstlane(t);          // uniform per wave
    const int* rp = rec + (size_t)ts * 8;
    int   i1 = rp[0];
    int   i2 = rp[1];
    int   r1 = rp[2];
    int   r2 = rp[3];
    float wa = __int_as_float(rp[4]);
    float wb = __int_as_float(rp[5]);

    // e is uniform across each wave -> wave-uniform branches
    int r = -1; float w = 0.0f;
    if (e == i1)      { r = r1; w = wa; }
    else if (e == i2) { r = r2; w = wb; }

    v4f v;
    v.x = (r == c0    ) ? w : 0.0f;
    v.y = (r == c0 + 1) ? w : 0.0f;
    v.z = (r == c0 + 2) ? w : 0.0f;
    v.w = (r == c0 + 3) ? w : 0.0f;
    v4f mk;
    mk.x = (v.x != 0.0f) ? 1.0f : 0.0f;
    mk.y = (v.y != 0.0f) ? 1.0f : 0.0f;
    mk.z = (v.z != 0.0f) ? 1.0f : 0.0f;
    mk.w = (v.w != 0.0f) ? 1.0f : 0.0f;

    // streaming: 1 GB write-once output must not pollute the 192 MB L2
    __builtin_nontemporal_store(v,  (v4f*)out + i);
    __builtin_nontemporal_store(mk, (v4f*)(out + N) + i);
}

// ---------------------------------------------------------------------------
extern "C" void kernel_launch(void* const* d_in, const int* in_sizes, int n_in,
                              void* d_out, int out_size, void* d_ws, size_t ws_size,
                              hipStream_t stream) {
    const float* in = (const float*)d_in[0];
    const int e = NUM_EXPERTS;
    const int s = in_sizes[0] / e;                       // 8192

    int cap = (int)(2.0 * s / e);                        // capacity_factor_eval = 2.0
    cap += cap % 2;
    if (cap < 4) cap = 4;                                // 256 for s=8192

    int*   rec = (int*)d_ws;                             // s * 8 ints (256 KB)
    float* out = (float*)d_out;

    long long    N    = (long long)s * e * cap;          // cb_weight elements
    unsigned int nvec = (unsigned int)(N >> 2);

    // Kernel 1: one wave32 per token, 8 waves per 256-thread block.
    int blocks1 = (s + 7) / 8;
    router_top2_kernel<<<blocks1, 256, 0, stream>>>(in, rec, s);

    // Kernel 2: single-block rank scan, async-pipelined, ballot-based.
    rank_scan_kernel<<<1, 256, 0, stream>>>(rec, s, cap);

    // Kernel 3: full-bandwidth fused fill+scatter (cb_weight + sec_mask).
    int tptShift = __builtin_ctz(16 * cap);              // threads per token = e*cap/4
    int cgShift  = __builtin_ctz(cap / 4);
    unsigned int blocks3 = (nvec + 255u) / 256u;
    fill_scatter_kernel<<<blocks3, 256, 0, stream>>>(rec, out, nvec, N, tptShift, cgShift);
}